// LDLLoss_18571438588271
// MI455X (gfx1250) — compile-verified
//
#include <hip/hip_runtime.h>
#include <stdint.h>

// ---------------------------------------------------------------------------
// LDL loss for MI455X (gfx1250).  Bandwidth-bound (~138 MB total traffic ->
// ~6us at 23.3 TB/s), so no WMMA; the CDNA5 path used here is the Tensor Data
// Mover (tensor_load_to_lds + s_wait_tensorcnt) for 70x70 halo-tile staging
// into LDS, wave32-sized blocks, and register-resident sliding-window box
// sums to keep LDS instruction count below the HBM roofline.
// ---------------------------------------------------------------------------

#define BATCH 16
#define CH 3
#define HH 512
#define WW 512
#define HW (HH * WW)              // 262144
#define PAD 3
#define HP (HH + 2 * PAD)         // 518
#define WP (WW + 2 * PAD)         // 518
#define TILE 64
#define TPAD (TILE + 2 * PAD)     // 70
#define TILES_X (WW / TILE)       // 8
#define TILES_PER_IMG (TILES_X * TILES_X)  // 64
#define P1_BLOCKS_PER_IMG 256
#define P1_PIX_PER_BLOCK 1024

typedef __attribute__((ext_vector_type(4))) unsigned int v4u;
typedef __attribute__((ext_vector_type(8))) int v8i;
typedef __attribute__((ext_vector_type(4))) int v4i;

// Hard requirement: this kernel is built around the CDNA5 Tensor Data Mover.
// Probe only on the DEVICE pass (AMDGPU builtins are not declared for host).
#if defined(__HIP_DEVICE_COMPILE__)
#if !__has_builtin(__builtin_amdgcn_tensor_load_to_lds)
#error "gfx1250 TDM builtin __builtin_amdgcn_tensor_load_to_lds not available on device"
#endif
#endif

// ---------------------------------------------------------------------------
// Pass 1: residual r = sum_c |out-gt|, write into padded image interior,
// accumulate per-block (sum, sumsq) partials for the global variance.
// ---------------------------------------------------------------------------
__global__ void k_residual(const float* __restrict__ outp,
                           const float* __restrict__ gtp,
                           float* __restrict__ padded,
                           float* __restrict__ p1s,
                           float* __restrict__ p1ss) {
  const int b   = blockIdx.x >> 8;        // / P1_BLOCKS_PER_IMG
  const int blk = blockIdx.x & 255;
  const int tid = threadIdx.x;
  const size_t ibase = (size_t)b * CH * HW;
  float s = 0.f, ss = 0.f;
#pragma unroll
  for (int i = 0; i < 4; ++i) {
    const int idx = blk * P1_PIX_PER_BLOCK + i * 256 + tid;
    const int h = idx >> 9;
    const int w = idx & (WW - 1);
    const float r = fabsf(outp[ibase + idx]          - gtp[ibase + idx]) +
                    fabsf(outp[ibase + HW + idx]     - gtp[ibase + HW + idx]) +
                    fabsf(outp[ibase + 2 * HW + idx] - gtp[ibase + 2 * HW + idx]);
    padded[(size_t)b * HP * WP + (size_t)(h + PAD) * WP + (w + PAD)] = r;
    s += r;
    ss += r * r;
  }
  __shared__ float rs[256];
  __shared__ float rq[256];
  rs[tid] = s;
  rq[tid] = ss;
  __syncthreads();
  for (int off = 128; off > 0; off >>= 1) {
    if (tid < off) { rs[tid] += rs[tid + off]; rq[tid] += rq[tid + off]; }
    __syncthreads();
  }
  if (tid == 0) { p1s[blockIdx.x] = rs[0]; p1ss[blockIdx.x] = rq[0]; }
}

// ---------------------------------------------------------------------------
// Pass 1b: materialize reflect padding so every TDM tile is fully in-bounds.
// (TDM out-of-bounds reads return zero, which would corrupt border windows.)
// ---------------------------------------------------------------------------
__device__ __forceinline__ int reflect_idx(int x) {
  return x < 0 ? -x : (x >= HH ? 2 * HH - 2 - x : x);
}

__global__ void k_padfill(float* __restrict__ padded) {
  const int b = blockIdx.x >> 3;
  const int chunk = blockIdx.x & 7;
  const int total = HP * WP;
  const int per = (total + 7) / 8;
  const int start = chunk * per;
  const int end = (start + per < total) ? (start + per) : total;
  float* img = padded + (size_t)b * HP * WP;
  for (int t = start + (int)threadIdx.x; t < end; t += 256) {
    const int i = t / WP;
    const int j = t - i * WP;
    if (i >= PAD && i < HP - PAD && j >= PAD && j < WP - PAD) continue;
    const int rh = reflect_idx(i - PAD);
    const int rw = reflect_idx(j - PAD);
    img[t] = img[(rh + PAD) * WP + (rw + PAD)];
  }
}

// ---------------------------------------------------------------------------
// Pass 2: one block per 64x64 tile. TDM loads the 70x70 halo tile into LDS,
// separable 7-tap box sums (sliding-window, register-resident) give the local
// unbiased variance; accumulate sum(pixel_w * r) per block.
// ---------------------------------------------------------------------------
__global__ void k_boxvar(const float* __restrict__ padded,
                         float* __restrict__ p2) {
  const int b    = blockIdx.x / TILES_PER_IMG;
  const int tile = blockIdx.x % TILES_PER_IMG;
  const int ty = tile / TILES_X;
  const int tx = tile % TILES_X;
  const int tid = threadIdx.x;

  // Single shared object => tr sits at LDS offset 0 (TDM lds_addr = 0).
  __shared__ float smem[TPAD * TPAD + 2 * TILE * TPAD + 256];
  float* tr  = smem;                       // 70x70 residual tile
  float* vs  = smem + TPAD * TPAD;         // 64x70 vertical sums
  float* vq  = vs + TILE * TPAD;           // 64x70 vertical sums of squares
  float* red = vq + TILE * TPAD;           // 256 reduction slots

  const float* gsrc =
      padded + (size_t)b * HP * WP + (size_t)(ty * TILE) * WP + (size_t)(tx * TILE);

#if defined(__HIP_DEVICE_COMPILE__)
  if ((tid >> 5) == 0) {  // wave 0 issues the DMA
    const uint64_t ga = (uint64_t)(uintptr_t)gsrc;
    // D# group 0: count=1, lds_addr=0, 57-bit global byte address, type=2.
    v4u g0 = {1u, 0u, (unsigned)(ga & 0xFFFFFFFFu),
              (unsigned)(((ga >> 32) & 0x01FFFFFFu) | (2u << 30))};
    // D# group 1: data_size=4B; tensor 518x518, row stride 518;
    // tile 70x70; tensor_dim1_stride = image plane (unused for 2D tile).
    const unsigned d0 = (unsigned)WP;            // tensor_dim0
    const unsigned d1 = (unsigned)HP;            // tensor_dim1
    const unsigned s0 = (unsigned)WP;            // tensor_dim0_stride
    const unsigned s1 = (unsigned)(HP * WP);     // tensor_dim1_stride
    v8i g1 = {(int)(2u << 16),
              (int)((d0 & 0xFFFFu) << 16),
              (int)(((d0 >> 16) & 0xFFFFu) | ((d1 & 0xFFFFu) << 16)),
              (int)(((d1 >> 16) & 0xFFFFu) | ((unsigned)TPAD << 16)),
              (int)TPAD,
              (int)s0,
              (int)((s1 & 0xFFFFu) << 16),
              (int)(s1 >> 16)};
    // Groups 2/3: higher dims length 1, tile dims 0 (unused).
    v4i g2 = {1, 1, 0, 0};
    v4i g3 = {0, 1 << 16, 0, 0};
#if __clang_major__ >= 23
    v8i g4 = {0, 0, 0, 0, 0, 0, 0, 0};
    __builtin_amdgcn_tensor_load_to_lds(g0, g1, g2, g3, g4, 0);
#else
    __builtin_amdgcn_tensor_load_to_lds(g0, g1, g2, g3, 0);
#endif
    __builtin_amdgcn_s_wait_tensorcnt(0);
  }
#else
  // Host-side parse-only body (never executed; device pass uses TDM above).
  for (int i = tid; i < TPAD * TPAD; i += 256)
    tr[i] = gsrc[(size_t)(i / TPAD) * WP + (i % TPAD)];
#endif
  __syncthreads();

  // Vertical 7-tap sums: 280 tasks, each a 16-row strip of one column.
  for (int task = tid; task < TPAD * 4; task += 256) {
    const int x = task % TPAD;
    const int y0 = (task / TPAD) * 16;
    float a[22];
#pragma unroll
    for (int k = 0; k < 22; ++k) a[k] = tr[(y0 + k) * TPAD + x];
    float s = a[0] + a[1] + a[2] + a[3] + a[4] + a[5] + a[6];
    float q = a[0] * a[0] + a[1] * a[1] + a[2] * a[2] + a[3] * a[3] +
              a[4] * a[4] + a[5] * a[5] + a[6] * a[6];
    vs[(y0)*TPAD + x] = s;
    vq[(y0)*TPAD + x] = q;
#pragma unroll
    for (int k = 1; k < 16; ++k) {
      s += a[k + 6] - a[k - 1];
      q += a[k + 6] * a[k + 6] - a[k - 1] * a[k - 1];
      vs[(y0 + k) * TPAD + x] = s;
      vq[(y0 + k) * TPAD + x] = q;
    }
  }
  __syncthreads();

  // Horizontal 7-tap sums + variance + weighted accumulation.
  // 256 tasks: each a 16-column strip of one row.
  float acc = 0.f;
  {
    const int y = tid & (TILE - 1);
    const int x0 = (tid >> 6) * 16;
    float sa[22], qa[22];
#pragma unroll
    for (int k = 0; k < 22; ++k) {
      sa[k] = vs[y * TPAD + x0 + k];
      qa[k] = vq[y * TPAD + x0 + k];
    }
    float s = sa[0] + sa[1] + sa[2] + sa[3] + sa[4] + sa[5] + sa[6];
    float q = qa[0] + qa[1] + qa[2] + qa[3] + qa[4] + qa[5] + qa[6];
#pragma unroll
    for (int k = 0; k < 16; ++k) {
      if (k > 0) {
        s += sa[k + 6] - sa[k - 1];
        q += qa[k + 6] - qa[k - 1];
      }
      const float pw = (q - s * s * (1.0f / 49.0f)) * (1.0f / 48.0f);
      acc += pw * tr[(y + PAD) * TPAD + (x0 + k + PAD)];
    }
  }
  red[tid] = acc;
  __syncthreads();
  for (int off = 128; off > 0; off >>= 1) {
    if (tid < off) red[tid] += red[tid + off];
    __syncthreads();
  }
  if (tid == 0) p2[blockIdx.x] = red[0];
}

// ---------------------------------------------------------------------------
// Pass 3: per-image global variance -> patch_w; combine with tile partials.
// ---------------------------------------------------------------------------
__global__ void k_final(const float* __restrict__ p1s,
                        const float* __restrict__ p1ss,
                        const float* __restrict__ p2,
                        float* __restrict__ outv) {
  __shared__ double vals[BATCH];
  const int t = threadIdx.x;
  if (t < BATCH) {
    double s = 0.0, q = 0.0;
    for (int i = 0; i < P1_BLOCKS_PER_IMG; ++i) {
      s += (double)p1s[t * P1_BLOCKS_PER_IMG + i];
      q += (double)p1ss[t * P1_BLOCKS_PER_IMG + i];
    }
    const double N = (double)HW;
    const double var = (q - s * s / N) / (N - 1.0);
    const double pw = pow(var, 0.2);
    double S = 0.0;
    for (int i = 0; i < TILES_PER_IMG; ++i) S += (double)p2[t * TILES_PER_IMG + i];
    vals[t] = pw * S;
  }
  __syncthreads();
  if (t == 0) {
    double L = 0.0;
    for (int b = 0; b < BATCH; ++b) L += vals[b];
    outv[0] = (float)(L / ((double)BATCH * CH * HH * WW));
  }
}

// ---------------------------------------------------------------------------
extern "C" void kernel_launch(void* const* d_in, const int* in_sizes, int n_in,
                              void* d_out, int out_size, void* d_ws, size_t ws_size,
                              hipStream_t stream) {
  const float* outp = (const float*)d_in[0];
  const float* gtp  = (const float*)d_in[1];

  float* ws = (float*)d_ws;
  float* padded = ws;                                         // B*518*518
  float* p1s  = padded + (size_t)BATCH * HP * WP;             // 4096
  float* p1ss = p1s + BATCH * P1_BLOCKS_PER_IMG;              // 4096
  float* p2   = p1ss + BATCH * P1_BLOCKS_PER_IMG;             // 1024

  k_residual<<<BATCH * P1_BLOCKS_PER_IMG, 256, 0, stream>>>(outp, gtp, padded, p1s, p1ss);
  k_padfill<<<BATCH * 8, 256, 0, stream>>>(padded);
  k_boxvar<<<BATCH * TILES_PER_IMG, 256, 0, stream>>>(padded, p2);
  k_final<<<1, 32, 0, stream>>>(p1s, p1ss, p2, (float*)d_out);
}